// SPGG_Qlearning_48095043781081
// MI455X (gfx1250) — compile-verified
//
#include <hip/hip_runtime.h>
#include <cstdint>

// ---------------------------------------------------------------------------
// SPGG + Q-learning, one fused pass on gfx1250 (MI455X).
//
// Memory-bound: ~832 MB HBM traffic -> ~36us floor @ 23.3 TB/s; compute is
// ~1 GFLOP (negligible). No matrix structure -> WMMA not applicable. CDNA5
// paths used:
//   * GLOBAL_LOAD_ASYNC_TO_LDS_B32 (ASYNCcnt) DMA of the halo'd type tile
//   * s_wait_asynccnt + split workgroup barriers between LDS stencil stages
//   * TH_NT temporal hints on all zero-reuse streams (Q in/out, ldir, lprob,
//     outputs) so the 192MB L2 is reserved for the type-halo overlap, the
//     only reused data.
// ---------------------------------------------------------------------------

#define LAT      4096
#define LMASK    4095
#define TILE     32
#define NTHREADS 256

// type tile: halo 3 (profit-of-neighbor needs type at radius 3)
#define TS   38
#define TSTR 39          // padded row stride (odd -> fewer bank conflicts)
// coop_num tile: halo 2
#define CN   36
#define CSTR 37
// profit tile: halo 1
#define PS   34
#define PSTR 35

#define R_OVER_5 0.98f   // R/5 with R = 4.9
#define INV_K    10.0f   // 1/K  with K = 0.1

typedef float v4f __attribute__((ext_vector_type(4)));

__device__ __forceinline__ void async_ld_b32(unsigned lds_off, const void* gaddr) {
    // GLOBAL_LOAD_ASYNC_TO_LDS_B32 (GV mode): per-lane LDS byte address in
    // vdst, per-lane 64-bit global address in vaddr pair. Tracked by ASYNCcnt.
    asm volatile("global_load_async_to_lds_b32 %0, %1, off"
                 :: "v"(lds_off), "v"(gaddr)
                 : "memory");
}

__device__ __forceinline__ void wait_asynccnt0() {
#if __has_builtin(__builtin_amdgcn_s_wait_asynccnt)
    __builtin_amdgcn_s_wait_asynccnt(0);
#else
    asm volatile("s_wait_asynccnt 0" ::: "memory");
#endif
}

__global__ __launch_bounds__(NTHREADS)
void spgg_qlearn_step(const int*   __restrict__ type_t,
                      const float* __restrict__ Q,        // [N,2,2] flat
                      const int*   __restrict__ ldir,
                      const float* __restrict__ lprob,
                      float*       __restrict__ outProfit,
                      float*       __restrict__ outType,
                      float*       __restrict__ outQ) {
    __shared__ int   sType[TS * TSTR];
    __shared__ float sCoop[CN * CSTR];
    __shared__ float sProf[PS * PSTR];

    const int t   = threadIdx.x;
    const int gi0 = blockIdx.y * TILE;
    const int gj0 = blockIdx.x * TILE;

    // ---- Stage 1: async DMA of type tile (+halo 3, periodic wrap) into LDS.
    // type_t is the ONLY reused stream (halo overlap between tiles) -> keep
    // it regular-temporal so L2 serves the overlap.
    for (int k = t; k < TS * TS; k += NTHREADS) {
        const int r  = k / TS;
        const int c  = k - r * TS;
        const int gi = (gi0 + r - 3) & LMASK;
        const int gj = (gj0 + c - 3) & LMASK;
        const int* gp = type_t + (size_t)gi * LAT + gj;
        const unsigned loff = (unsigned)(uintptr_t)(&sType[r * TSTR + c]);
        async_ld_b32(loff, gp);
    }
    wait_asynccnt0();       // this wave's async LDS writes landed
    __syncthreads();        // all waves' writes visible

    // ---- Stage 2: coop_num = 5-pt stencil of cooperator flags (halo 2)
    for (int k = t; k < CN * CN; k += NTHREADS) {
        const int r = k / CN;
        const int c = k - r * CN;
        const int b = (r + 1) * TSTR + (c + 1);   // type-tile coords = coop+1
        const int cn = (sType[b]        == 1) + (sType[b - TSTR] == 1) +
                       (sType[b + TSTR] == 1) + (sType[b - 1]    == 1) +
                       (sType[b + 1]    == 1);
        sCoop[r * CSTR + c] = (float)cn;
    }
    __syncthreads();

    // ---- Stage 3: profit = (R/5)*nsum(coop_num) - 5*[type==1]   (halo 1)
    for (int k = t; k < PS * PS; k += NTHREADS) {
        const int r = k / PS;
        const int c = k - r * PS;
        const int b = (r + 1) * CSTR + (c + 1);   // coop coords = profit+1
        const float s = sCoop[b] + sCoop[b - CSTR] + sCoop[b + CSTR] +
                        sCoop[b - 1] + sCoop[b + 1];
        const int ty = sType[(r + 2) * TSTR + (c + 2)];
        sProf[r * PSTR + c] = s * R_OVER_5 - ((ty == 1) ? 5.0f : 0.0f);
    }
    __syncthreads();

    // ---- Stage 4: Fermi imitation + Q-table update.
    // All global traffic here is read-once/write-once -> non-temporal.
    for (int k = t; k < TILE * TILE; k += NTHREADS) {
        const int li = k >> 5;
        const int lj = k & 31;
        const int gi = gi0 + li;
        const int gj = gj0 + lj;
        const size_t cell = (size_t)gi * LAT + gj;

        const float p    = sProf[(li + 1) * PSTR + (lj + 1)];
        const int   tOld = sType[(li + 3) * TSTR + (lj + 3)];

        // dirs: 0 -> (0,-1)  1 -> (0,+1)  2 -> (-1,0)  3 -> (+1,0)
        const int d  = __builtin_nontemporal_load(ldir + cell);
        const int di = (d == 2) ? -1 : ((d == 3) ? 1 : 0);
        const int dj = (d == 0) ? -1 : ((d == 1) ? 1 : 0);

        const float pn  = sProf[(li + 1 + di) * PSTR + (lj + 1 + dj)];
        const float w   = 1.0f / (1.0f + expf((p - pn) * INV_K));
        const int   tNb = sType[(li + 3 + di) * TSTR + (lj + 3 + dj)];
        const float lpv = __builtin_nontemporal_load(lprob + cell);
        const int  tNew = (lpv <= w) ? tNb : tOld;

        // Q gather (old table) -> scalar update at [A,B] -> float4 scatter
        const v4f q = __builtin_nontemporal_load((const v4f*)Q + cell); // b128 NT
        const int A = tOld, B = tNew;
        const float qB0  = B ? q.z : q.x;                // Q[cell,B,0]
        const float qB1  = B ? q.w : q.y;                // Q[cell,B,1]
        const float mx   = fmaxf(qB0, qB1);
        const float qab  = A ? (B ? q.w : q.z) : (B ? q.y : q.x);
        const float upd  = 0.2f * qab + 0.8f * (p + 0.8f * mx);  // eta=gamma=0.8
        const int   sel  = A * 2 + B;

        v4f qo;
        qo.x = (sel == 0) ? upd : q.x;
        qo.y = (sel == 1) ? upd : q.y;
        qo.z = (sel == 2) ? upd : q.z;
        qo.w = (sel == 3) ? upd : q.w;

        __builtin_nontemporal_store(qo, (v4f*)outQ + cell);     // b128 NT
        __builtin_nontemporal_store(p, outProfit + cell);
        __builtin_nontemporal_store((float)tNew, outType + cell);
    }
}

extern "C" void kernel_launch(void* const* d_in, const int* in_sizes, int n_in,
                              void* d_out, int out_size, void* d_ws, size_t ws_size,
                              hipStream_t stream) {
    const int*   type_t = (const int*)d_in[0];
    const float* Q      = (const float*)d_in[1];
    const int*   ldir   = (const int*)d_in[2];
    const float* lprob  = (const float*)d_in[3];

    float* out = (float*)d_out;
    const size_t N = (size_t)LAT * LAT;

    dim3 grid(LAT / TILE, LAT / TILE);   // 128 x 128 tiles
    spgg_qlearn_step<<<grid, NTHREADS, 0, stream>>>(
        type_t, Q, ldir, lprob,
        out,            // profit
        out + N,        // type_t1 (as float)
        out + 2 * N);   // Q_new
}